// Correlation1dCost_26482768347744
// MI455X (gfx1250) — compile-verified
//
#include <hip/hip_runtime.h>

// Problem constants (from reference): B=8, C=128, H=128, W=256, MAX_DISP=48
#define NB 8
#define NC 128
#define NH 128
#define NW 256
#define ND 48

#define XW 128      // x-columns handled per block (half of a row)
#define BW 176      // feat2 columns staged per block = XW + 48 halo
#define SA2 136     // As LDS row stride (floats): 2*136 % 64 = 16 -> bank-disjoint halves
#define SB2 176     // Bs LDS row stride (floats): 2*176 % 64 = 32 -> bank-disjoint halves
#define AS_FLOATS (NC * SA2)                      // 17408
#define BS_FLOATS (NC * SB2)                      // 22528
#define SMEM_BYTES ((AS_FLOATS + BS_FLOATS) * 4)  // 159744 B -> 2 blocks per 320KB WGP

typedef __attribute__((ext_vector_type(2))) float v2f;
typedef __attribute__((ext_vector_type(8))) float v8f;
typedef __attribute__((ext_vector_type(4))) unsigned int v4u;
typedef __attribute__((ext_vector_type(8))) int v8i;
typedef __attribute__((ext_vector_type(4))) int v4i;

// Issue a TDM 2-D tile load: global (tile_w x rows, row stride row_stride_elts f32)
// -> LDS at lds_off, with optional per-row LDS padding (pad codes per ISA 8.4).
// D# group0/group1 bitfields per CDNA5 ISA section 8.3/8.4; groups 2/3 zero (2-D).
static __device__ __forceinline__ void tdm_load_2d(unsigned lds_off, const void* gptr,
                                                   unsigned tile_w, unsigned rows,
                                                   unsigned row_stride_elts, int pad_en,
                                                   unsigned pad_interval_code,
                                                   unsigned pad_amount_code) {
  unsigned long long ga = (unsigned long long)gptr;
  v4u g0;
  g0.x = 1u;                                                    // count=1, user, no gather
  g0.y = lds_off;                                               // lds_addr (bytes)
  g0.z = (unsigned)ga;                                          // global_addr[31:0]
  g0.w = (unsigned)((ga >> 32) & 0x01FFFFFFull) | 0x80000000u;  // addr[56:32] | type=2
  v8i g1;
  g1[0] = (int)((2u << 16) |                      // data_size = 4 bytes
                ((unsigned)pad_en << 20) |        // pad_enable
                (pad_interval_code << 22) |       // pad interval (2^(c+1) dwords)
                (pad_amount_code << 25));         // pad amount (c+1 dwords)
  g1[1] = (int)((tile_w & 0xFFFFu) << 16);        // tensor_dim0[15:0]
  g1[2] = (int)((tile_w >> 16) | ((rows & 0xFFFFu) << 16));  // dim0 hi | tensor_dim1 lo
  g1[3] = (int)((rows >> 16) | (tile_w << 16));   // dim1 hi | tile_dim0
  g1[4] = (int)rows;                              // tile_dim1 (tile_dim2 = 0)
  g1[5] = (int)row_stride_elts;                   // tensor_dim0_stride[31:0]
  g1[6] = 0;                                      // stride hi / dim1_stride (unused, 2-D)
  g1[7] = 0;
  v4i z4 = {0, 0, 0, 0};
  v8i z8 = {0, 0, 0, 0, 0, 0, 0, 0};
  // amdgpu-toolchain (clang-23) form: 6 args, trailing int32x8 before cpol.
  __builtin_amdgcn_tensor_load_to_lds(g0, g1, z4, z4, z8, 0);
}

// One block per (b, y, half-row). 256 threads = 8 wave32's; wave w owns local
// x-tile [16w, 16w+16). TDM stages feat1 (128x128) and feat2 (176x128 incl. 48-col
// halo; left edge zero-padded) into LDS; each wave then computes 4 WMMA 16x16 Gram
// tiles, 32 x V_WMMA_F32_16X16X4_F32 each, over C=128.
__global__ __launch_bounds__(256)
void corr1d_wmma_tdm_kernel(const float* __restrict__ f1,
                            const float* __restrict__ f2,
                            float* __restrict__ out) {
  extern __shared__ float smem[];
  float* As = smem;              // [NC][SA2], col = local x in [0,128)
  float* Bs = smem + AS_FLOATS;  // [NC][SB2], col = padded x' (global x' - (X0-48))

  const int tid = threadIdx.x;
  const int halfBlk = blockIdx.x & 1;
  const int bh = blockIdx.x >> 1;
  const int b = bh >> 7;
  const int y = bh & (NH - 1);
  const int X0 = halfBlk * XW;

  const size_t hw = (size_t)NH * NW;
  const size_t chw = (size_t)NC * hw;
  const float* p1 = f1 + (size_t)b * chw + (size_t)y * NW + X0;
  const float* p2 = f2 + (size_t)b * chw + (size_t)y * NW;

  const unsigned AsOff = (unsigned)(unsigned long long)(As);
  const unsigned BsOff = AsOff + AS_FLOATS * 4;

  // Wave 0 issues the TDM tile loads; EXEC is full (uniform branch).
  if ((tid >> 5) == 0) {
    // feat1 tile: 128 cols x 128 rows, LDS row stride 136 = 128 data + 8-dword pad
    // (pad_interval code 6 = 128 dwords, pad_amount code 7 = 8 dwords).
    tdm_load_2d(AsOff, p1, XW, NC, (unsigned)hw, 1, 6, 7);
    if (X0 == 0) {
      // Left edge: only x' in [0,128) exists; land it at column 48, pad 48 dwords
      // per 128-dword row so the row stride stays 176.
      tdm_load_2d(BsOff + 48 * 4, p2, XW, NC, (unsigned)hw, 1, 6, 47);
    } else {
      // Interior: full 176-wide halo tile starting at x' = X0 - 48, stride 176 (no pad).
      tdm_load_2d(BsOff, p2 + (X0 - 48), BW, NC, (unsigned)hw, 0, 0, 0);
    }
    __builtin_amdgcn_s_wait_tensorcnt(0);
  }
  // Zero the 48-column left pad for the edge block (disjoint from TDM's region).
  if (X0 == 0) {
    for (int i = tid; i < NC * 12; i += 256) {
      int c = i / 12, q = i - c * 12;
      *(float4*)&Bs[c * SB2 + q * 4] = make_float4(0.f, 0.f, 0.f, 0.f);
    }
  }
  __syncthreads();

  const int lane = tid & 31;
  const int wave = tid >> 5;
  const int half = lane >> 4;  // K-half selector within fragments
  const int mn = lane & 15;    // M (for A) / N (for B) within the tile
  const int x0l = wave * 16;   // local x-tile base

  // Preload all 32 A fragments (K-steps): 64 VGPRs, reused across the 4 B tiles.
  float a0[32], a1[32];
#pragma unroll
  for (int k = 0; k < 32; ++k) {
    const int row = 4 * k + 2 * half;
    a0[k] = As[row * SA2 + x0l + mn];
    a1[k] = As[(row + 1) * SA2 + x0l + mn];
  }

  const size_t obase = (size_t)b * ND * hw + (size_t)y * NW;

  // Band out[d, x] needs local Bs columns [x0l+1, x0l+63]:
  // exactly 4 aligned 16-wide tiles starting at xq0 = x0l + 16j.
#pragma unroll
  for (int j = 0; j < 4; ++j) {
    const int xq0 = x0l + 16 * j;
    v8f acc = {0.f, 0.f, 0.f, 0.f, 0.f, 0.f, 0.f, 0.f};
#pragma unroll
    for (int k = 0; k < 32; ++k) {
      const int row = 4 * k + 2 * half;
      v2f a;
      a.x = a0[k];
      a.y = a1[k];
      v2f bb;
      bb.x = Bs[row * SB2 + xq0 + mn];
      bb.y = Bs[(row + 1) * SB2 + xq0 + mn];
      // D(f32 16x16) += A(16x4 f32) * B(4x16 f32)
      acc = __builtin_amdgcn_wmma_f32_16x16x4_f32(
          false, a, false, bb, (short)0, acc, false, false);
    }
    // D[m][n] -> out[d = 16j + n - m - 1, x = X0 + x0l + m], leaky ReLU 0.1.
    // Zero-halo columns compute to 0 -> leaky(0) = 0, matching the reference mask.
#pragma unroll
    for (int r = 0; r < 8; ++r) {
      const int m = r + 8 * half;  // C/D layout: VGPR r holds M=r (lanes 0-15), M=r+8 (16-31)
      const int d = 16 * j + mn - m - 1;
      if (d >= 0 && d < ND) {
        float v = acc[r];
        v = (v > 0.f) ? v : 0.1f * v;
        out[obase + (size_t)d * hw + X0 + x0l + m] = v;
      }
    }
  }
}

extern "C" void kernel_launch(void* const* d_in, const int* in_sizes, int n_in,
                              void* d_out, int out_size, void* d_ws, size_t ws_size,
                              hipStream_t stream) {
  (void)in_sizes; (void)n_in; (void)out_size; (void)d_ws; (void)ws_size;
  const float* feat1 = (const float*)d_in[0];
  const float* feat2 = (const float*)d_in[1];
  float* out = (float*)d_out;

  static_assert(SMEM_BYTES <= 160 * 1024, "need 2 blocks per 320KB WGP");
  (void)hipFuncSetAttribute((const void*)corr1d_wmma_tdm_kernel,
                            hipFuncAttributeMaxDynamicSharedMemorySize, SMEM_BYTES);

  dim3 grid(NB * NH * 2);  // 2048 blocks: (b, y, half-row)
  dim3 block(256);         // 8 wave32's
  corr1d_wmma_tdm_kernel<<<grid, block, SMEM_BYTES, stream>>>(feat1, feat2, out);
}